// TransportLpLoss_17454747091270
// MI455X (gfx1250) — compile-verified
//
#include <hip/hip_runtime.h>
#include <hip/hip_bf16.h>

typedef float v2f __attribute__((ext_vector_type(2)));
typedef float v8f __attribute__((ext_vector_type(8)));

#define NPTS 1024
#define NSAMP 8
#define NEGF (-1e30f)
#define MAXR 5000

// ds_swizzle group-of-32 XOR pattern: offset[14:10]=xor, [9:5]=or=0, [4:0]=and=0x1f
#define SWIZ_F(v, m) __int_as_float(__builtin_amdgcn_ds_swizzle(__float_as_int(v), (0x1f | ((m) << 10))))
#define SWIZ_I(v, m) __builtin_amdgcn_ds_swizzle((v), (0x1f | ((m) << 10)))

// one merge step of the per-half top-2 reduction (pure cndmask, no branches)
#define MERGE_STEP(M)                                                          \
  do {                                                                         \
    _Pragma("unroll")                                                          \
    for (int k = 0; k < 8; ++k) {                                              \
      const float o1 = SWIZ_F(b1[k], M);                                       \
      const float o2 = SWIZ_F(b2[k], M);                                       \
      const int   oj = SWIZ_I(bj[k], M);                                       \
      const int   gt = (int)(o1 > b1[k]);                                      \
      const int   eq = (int)(o1 == b1[k]);                                     \
      const int   lo = (int)((unsigned)oj < (unsigned)bj[k]);                  \
      const bool  take = (gt | (eq & lo)) != 0;                                \
      const float secA = fmaxf(b1[k], o2);                                     \
      const float secB = fmaxf(b2[k], o1);                                     \
      b2[k] = take ? secA : secB;                                              \
      b1[k] = take ? o1 : b1[k];                                               \
      bj[k] = take ? oj : bj[k];                                               \
    }                                                                          \
  } while (0)

__global__ __launch_bounds__(1024) void tlp_auction_kernel(
    const float* __restrict__ x, const float* __restrict__ y,
    float* __restrict__ partial)
{
  __shared__ float xs[NPTS], ys[NPTS], price[NPTS];
  __shared__ float bcol[2 * NPTS];          // [j]=y_j ; [1024+j]=-(j^2+y_j^2+price_j)
  __shared__ float tv1[NPTS], tv2[NPTS];
  __shared__ int   tjs[NPTS];
  __shared__ int   bb[NPTS], winner[NPTS], assg[NPTS], owner[NPTS];
  __shared__ int   cnt;
  __shared__ float accum;

  const int s    = blockIdx.x;       // sample
  const int tid  = threadIdx.x;      // 0..1023 (person id / object id)
  const int lane = tid & 31;         // wave32 lane
  const int wave = tid >> 5;         // 0..31

  xs[tid]    = x[s * NPTS + tid];
  ys[tid]    = y[s * NPTS + tid];
  price[tid] = 0.0f;
  __syncthreads();

  const int  r16    = lane & 15;
  const bool hiHalf = (lane >= 16);
  const int  bsel   = hiHalf ? NPTS : 0;   // which half of bcol this lane reads

  for (int lev = 0; lev < 4; ++lev) {
    const float eps = (lev == 0) ? 1000.0f : (lev == 1) ? 100.0f
                    : (lev == 2) ? 10.0f   : 1.0f;
    assg[tid]  = -1;
    owner[tid] = -1;
    if (tid == 0) cnt = NPTS;
    __syncthreads();

    for (int round = 0; round < MAXR; ++round) {
      if (cnt == 0) break;                       // uniform: LDS read after barrier
      bb[tid]     = __float_as_int(NEGF);
      winner[tid] = -1;
      {
        const float jf = (float)tid;
        const float yj = ys[tid];
        bcol[tid]        = yj;
        bcol[NPTS + tid] = -(jf * jf + yj * yj + price[tid]);
      }
      __syncthreads();

      // ---------- top-2 over objects per person, 16x16 tiles via WMMA ------
      // val[i,j] = 2*i*j + 2*x_i*y_j - (i^2 + x_i^2) - (j^2 + y_j^2 + price_j)
      //   A row i = [2i, 2x_i, -(i^2+x_i^2), 1]   (K=0,1 lanes 0-15; K=2,3 lanes 16-31)
      //   B col j = [j,  y_j,  1, -(j^2+y_j^2+price_j)]
      for (int gi = 0; gi < 2; ++gi) {           // 32 waves * 2 = 64 person-groups
        const int rbase = (wave * 2 + gi) * 16;
        const int prow  = rbase + r16;

        // skip fully-assigned groups (wave-uniform branch: EXEC stays all-ones)
        if (__ballot(assg[prow] < 0) == 0ull) continue;

        const float pif = (float)prow;
        const float xi  = xs[prow];
        v2f A;
        A.x = hiHalf ? -(pif * pif + xi * xi) : 2.0f * pif;
        A.y = hiHalf ? 1.0f                   : 2.0f * xi;

        float b1[8], b2[8];
        int   bj[8];
        #pragma unroll
        for (int k = 0; k < 8; ++k) { b1[k] = NEGF; b2[k] = NEGF; bj[k] = -1; }

        // software-pipelined bcol prefetch (wrap keeps last prefetch in-range)
        int   c      = r16;
        float byNext = bcol[bsel + c];
        float cf     = (float)r16;
        #pragma unroll 2
        for (int t = 0; t < NPTS / 16; ++t) {
          const float by    = byNext;
          const int   cCur  = c;
          c = (c + 16) & (NPTS - 1);
          byNext = bcol[bsel + c];               // next tile's broadcast value
          v2f B;
          B.x = hiHalf ? 1.0f : cf;
          B.y = by;
          v8f d = {};
          d = __builtin_amdgcn_wmma_f32_16x16x4_f32(false, A, false, B,
                                                    (short)0, d, false, false);
          // D: VGPR k, lane<16 -> (M=k, N=lane); lane>=16 -> (M=k+8, N=lane-16)
          #pragma unroll
          for (int k = 0; k < 8; ++k) {          // cmp+select top-2 (no canonicalize)
            const float v  = d[k];
            const bool  g1 = v > b1[k];
            const bool  g2 = v > b2[k];
            const float t2 = g2 ? v : b2[k];
            b2[k] = g1 ? b1[k] : t2;
            bj[k] = g1 ? cCur  : bj[k];
            b1[k] = g1 ? v     : b1[k];
          }
          cf += 16.0f;
        }
        // merge per-lane top-2 across the 16-lane half (xor of bits 0..3)
        MERGE_STEP(1);
        MERGE_STEP(2);
        MERGE_STEP(4);
        MERGE_STEP(8);
        if (r16 == 0) {                          // lanes 0 & 16 publish 8 rows each
          const int roff = hiHalf ? 8 : 0;
          #pragma unroll
          for (int k = 0; k < 8; ++k) {
            const int p = rbase + roff + k;
            tv1[p] = b1[k]; tv2[p] = b2[k]; tjs[p] = bj[k];
          }
        }
      }
      __syncthreads();

      // ---------- bidding: scatter-max of bids per object -------------------
      float mybid = NEGF;
      int   myj   = -1;
      if (assg[tid] < 0) {
        myj   = tjs[tid];
        mybid = tv1[tid] - tv2[tid] + eps + price[myj];  // always > 0
        atomicMax(&bb[myj], __float_as_int(mybid));      // int cmp == float cmp (>0)
      }
      __syncthreads();
      // ---------- winner: max person id among max-bidders -------------------
      if (myj >= 0 && __float_as_int(mybid) >= bb[myj]) {
        atomicMax(&winner[myj], tid);
      }
      __syncthreads();
      // ---------- price update + evict previous owners ----------------------
      const float bbv = __int_as_float(bb[tid]);
      const bool  has = bbv > NEGF * 0.5f;
      if (has) {
        price[tid] = bbv;
        const int old = owner[tid];
        if (old >= 0) assg[old] = -1;
      }
      __syncthreads();
      // ---------- assign winners --------------------------------------------
      if (has) {
        const int w = winner[tid];
        assg[w]    = tid;
        owner[tid] = w;
      }
      __syncthreads();
      // ---------- recount unassigned ----------------------------------------
      if (tid == 0) cnt = 0;
      __syncthreads();
      const unsigned long long ball = __ballot(assg[tid] < 0);  // wave32 mask
      if (lane == 0) atomicAdd(&cnt, (int)__popcll(ball));
      __syncthreads();
    }
    __syncthreads();
  }

  // ---------- loss: sum_i (i - sigma)^2 + (x_i - y_sigma)^2 -----------------
  const int sig = assg[tid];
  float l = 0.0f;
  if (sig >= 0) {
    const float dd = (float)tid - (float)sig;
    const float dv = xs[tid] - ys[sig];
    l = dd * dd + dv * dv;
  }
  if (tid == 0) accum = 0.0f;
  __syncthreads();
  #pragma unroll
  for (int m = 16; m >= 1; m >>= 1) l += __shfl_xor(l, m, 32);
  if (lane == 0) atomicAdd(&accum, l);
  __syncthreads();
  if (tid == 0) partial[s] = accum;
}

__global__ void tlp_finalize_kernel(const float* __restrict__ partial,
                                    float* __restrict__ out)
{
  if (threadIdx.x == 0) {
    float acc = 0.0f;
    #pragma unroll
    for (int i = 0; i < NSAMP; ++i) acc += partial[i];
    out[0] = acc;                                 // deterministic fixed-order sum
  }
}

extern "C" void kernel_launch(void* const* d_in, const int* in_sizes, int n_in,
                              void* d_out, int out_size, void* d_ws, size_t ws_size,
                              hipStream_t stream) {
  (void)in_sizes; (void)n_in; (void)out_size; (void)ws_size;
  const float* x = (const float*)d_in[0];
  const float* y = (const float*)d_in[1];
  float* partial = (float*)d_ws;                  // 8 floats of scratch
  float* out     = (float*)d_out;

  tlp_auction_kernel<<<NSAMP, NPTS, 0, stream>>>(x, y, partial);
  tlp_finalize_kernel<<<1, 32, 0, stream>>>(partial, out);
}